// MatchOneLayer_79405355368450
// MI455X (gfx1250) — compile-verified
//
#include <hip/hip_runtime.h>
#include <hip/hip_bf16.h>
#include <math.h>

// ---------------------------------------------------------------------------
// Problem constants
// ---------------------------------------------------------------------------
#define NMODELS 8
#define N 2048                  // INPUT_SIZE == OUTPUT_SIZE == 2048
#define SINKHORN_ITERS 100
#define PDIST_EPS 1e-6f

typedef float v2f __attribute__((ext_vector_type(2)));
typedef float v8f __attribute__((ext_vector_type(8)));

// ---------------------------------------------------------------------------
// wave32 reductions
// ---------------------------------------------------------------------------
__device__ __forceinline__ float wave_max(float v) {
    #pragma unroll
    for (int o = 16; o > 0; o >>= 1) v = fmaxf(v, __shfl_xor(v, o, 32));
    return v;
}
__device__ __forceinline__ float wave_sum(float v) {
    #pragma unroll
    for (int o = 16; o > 0; o >>= 1) v += __shfl_xor(v, o, 32);
    return v;
}

// ---------------------------------------------------------------------------
// Sinkhorn: row logsumexp normalization (axis=2, contiguous rows).
// One wave per row; entire row cached in registers (16 x float4 per lane).
// grid: NMODELS*N/8 blocks of 256 (8 waves).
// ---------------------------------------------------------------------------
__global__ __launch_bounds__(256) void row_lse_kernel(float* __restrict__ p) {
    const int row  = blockIdx.x * 8 + (threadIdx.x >> 5);   // 0 .. 8*2048-1
    const int lane = threadIdx.x & 31;
    float* r = p + (size_t)row * N;
    const float4* r4 = (const float4*)r;

    float4 v[16];
    #pragma unroll
    for (int t = 0; t < 16; ++t) v[t] = r4[lane + 32 * t];

    float m = -__builtin_inff();
    #pragma unroll
    for (int t = 0; t < 16; ++t) {
        m = fmaxf(m, fmaxf(fmaxf(v[t].x, v[t].y), fmaxf(v[t].z, v[t].w)));
    }
    m = wave_max(m);

    float s = 0.f;
    #pragma unroll
    for (int t = 0; t < 16; ++t) {
        s += __expf(v[t].x - m) + __expf(v[t].y - m) +
             __expf(v[t].z - m) + __expf(v[t].w - m);
    }
    s = wave_sum(s);
    const float lse = m + __logf(s);

    float4* w4 = (float4*)r;
    #pragma unroll
    for (int t = 0; t < 16; ++t) {
        float4 o;
        o.x = v[t].x - lse; o.y = v[t].y - lse;
        o.z = v[t].z - lse; o.w = v[t].w - lse;
        w4[lane + 32 * t] = o;
    }
}

// ---------------------------------------------------------------------------
// Sinkhorn: column logsumexp normalization (axis=1).
// One thread per column, streaming online logsumexp (coalesced across lanes).
// grid: NMODELS*N/256 = 64 blocks of 256.
// ---------------------------------------------------------------------------
__global__ __launch_bounds__(256) void col_lse_kernel(float* __restrict__ p) {
    const int g     = blockIdx.x * 256 + threadIdx.x;   // 0 .. 8*2048-1
    const int model = g >> 11;
    const int c     = g & (N - 1);
    float* base = p + (size_t)model * N * N + c;

    float m = -__builtin_inff();
    float s = 0.f;
    for (int r = 0; r < N; ++r) {
        const float x  = base[(size_t)r * N];
        const float nm = fmaxf(m, x);
        s = s * __expf(m - nm) + __expf(x - nm);
        m = nm;
    }
    const float lse = m + __logf(s);
    for (int r = 0; r < N; ++r) base[(size_t)r * N] -= lse;
}

// ---------------------------------------------------------------------------
// row_sums = sum(exp(p), axis=2); accumulate (s-1)^2 into out[3]
// ---------------------------------------------------------------------------
__global__ __launch_bounds__(256) void row_sums_kernel(const float* __restrict__ p,
                                                       float* __restrict__ out) {
    const int row  = blockIdx.x * 8 + (threadIdx.x >> 5);
    const int lane = threadIdx.x & 31;
    const float4* r4 = (const float4*)(p + (size_t)row * N);
    float s = 0.f;
    #pragma unroll
    for (int t = 0; t < 16; ++t) {
        float4 v = r4[lane + 32 * t];
        s += __expf(v.x) + __expf(v.y) + __expf(v.z) + __expf(v.w);
    }
    s = wave_sum(s);
    if (lane == 0) {
        const float d = s - 1.0f;
        atomicAdd(&out[3], d * d);
    }
}

// ---------------------------------------------------------------------------
// col_sums = sum(exp(p), axis=1); accumulate (s-1)^2 into out[2]
// ---------------------------------------------------------------------------
__global__ __launch_bounds__(256) void col_sums_kernel(const float* __restrict__ p,
                                                       float* __restrict__ out) {
    const int g     = blockIdx.x * 256 + threadIdx.x;
    const int model = g >> 11;
    const int c     = g & (N - 1);
    const float* base = p + (size_t)model * N * N + c;
    float s = 0.f;
    for (int r = 0; r < N; ++r) s += __expf(base[(size_t)r * N]);
    const float d = s - 1.0f;
    atomicAdd(&out[2], d * d);
}

// ---------------------------------------------------------------------------
// WMMA fp32 GEMM: T[model] = W[model] @ exp(p[model]),   model in {0,1}
// Block = 256 threads (8 waves, 2x4 wave grid) computes 64(M) x 128(N).
// Each wave: 32x32 output = 2x2 tiles of V_WMMA_F32_16X16X4_F32.
// exp() fused into the B-tile LDS staging.
// ---------------------------------------------------------------------------
#define BM 64
#define BN 128
#define BK 32
#define AS_STRIDE 34   // even -> 8B-aligned b64 frag loads; conflict-free
#define BS_STRIDE 132  // keeps the two lane-halves on disjoint banks

__global__ __launch_bounds__(256) void gemm_wmma_kernel(const float* __restrict__ W,
                                                        const float* __restrict__ P,
                                                        float* __restrict__ T) {
    __shared__ float As[BM * AS_STRIDE];   // 8704 B
    __shared__ float Bs[BK * BS_STRIDE];   // 16896 B

    const int tid  = threadIdx.x;
    const int wave = tid >> 5;
    const int lane = tid & 31;
    const int wm   = wave >> 2;        // 0..1
    const int wn   = wave & 3;         // 0..3
    const int lh   = lane >> 4;        // lane half (K-pair select)
    const int ll   = lane & 15;

    const size_t moff = (size_t)blockIdx.z * N * N;
    const float* A = W + moff;
    const float* B = P + moff;
    float*       C = T + moff;

    const int m0 = blockIdx.y * BM;
    const int n0 = blockIdx.x * BN;

    v8f acc[2][2] = {};

    for (int k0 = 0; k0 < N; k0 += BK) {
        // --- stage A tile (64 x 32): 1024 float2, 4 per thread, coalesced ---
        #pragma unroll
        for (int q = 0; q < 4; ++q) {
            const int f2  = tid + 256 * q;          // 0..1023
            const int row = f2 >> 4;                // 16 float2 per row
            const int col = (f2 & 15) << 1;
            v2f a = *(const v2f*)(A + (size_t)(m0 + row) * N + k0 + col);
            *(v2f*)(As + row * AS_STRIDE + col) = a;
        }
        // --- stage B tile (32 x 128) with fused exp: 1024 float4 ---
        #pragma unroll
        for (int q = 0; q < 4; ++q) {
            const int f4  = tid + 256 * q;          // 0..1023
            const int row = f4 >> 5;                // 32 float4 per row
            const int col = (f4 & 31) << 2;
            float4 b = *(const float4*)(B + (size_t)(k0 + row) * N + n0 + col);
            float4 e;
            e.x = __expf(b.x); e.y = __expf(b.y);
            e.z = __expf(b.z); e.w = __expf(b.w);
            *(float4*)(Bs + row * BS_STRIDE + col) = e;
        }
        __syncthreads();

        #pragma unroll
        for (int kk = 0; kk < BK; kk += 4) {
            // A fragment: lanes 0-15 hold K=kk,kk+1 ; lanes 16-31 hold K=kk+2,kk+3
            v2f af[2];
            #pragma unroll
            for (int tm = 0; tm < 2; ++tm) {
                const int row = wm * 32 + tm * 16 + ll;
                af[tm] = *(const v2f*)(As + row * AS_STRIDE + kk + lh * 2);
            }
            // B fragment: vgpr0=K row (kk+lh*2), vgpr1=K row (kk+lh*2+1)
            v2f bf[2];
            #pragma unroll
            for (int tn = 0; tn < 2; ++tn) {
                const int col = wn * 32 + tn * 16 + ll;
                const int kr  = kk + lh * 2;
                bf[tn].x = Bs[kr * BS_STRIDE + col];
                bf[tn].y = Bs[(kr + 1) * BS_STRIDE + col];
            }
            #pragma unroll
            for (int tm = 0; tm < 2; ++tm)
                #pragma unroll
                for (int tn = 0; tn < 2; ++tn)
                    acc[tm][tn] = __builtin_amdgcn_wmma_f32_16x16x4_f32(
                        false, af[tm], false, bf[tn],
                        (short)0, acc[tm][tn], false, false);
        }
        __syncthreads();
    }

    // epilogue: C/D layout -> vgpr r holds M = r + lh*8, N = ll
    #pragma unroll
    for (int tm = 0; tm < 2; ++tm)
        #pragma unroll
        for (int tn = 0; tn < 2; ++tn)
            #pragma unroll
            for (int r = 0; r < 8; ++r) {
                const int row = m0 + wm * 32 + tm * 16 + lh * 8 + r;
                const int col = n0 + wn * 32 + tn * 16 + ll;
                C[(size_t)row * N + col] = acc[tm][tn][r];
            }
}

// ---------------------------------------------------------------------------
// Pairwise-distance loss: per output column o,
//   loss_o = sqrt( sum_i (T0[i,o] - T1[i,o] + eps)^2 )
// Stage 1: partial sums over i-chunks into colacc[o].
// ---------------------------------------------------------------------------
__global__ __launch_bounds__(256) void loss_partial_kernel(const float* __restrict__ T,
                                                           float* __restrict__ colacc) {
    const int o  = blockIdx.x * 256 + threadIdx.x;   // column 0..2047
    const int i0 = blockIdx.y * 128;
    const float* T0 = T;
    const float* T1 = T + (size_t)N * N;
    float s = 0.f;
    for (int i = i0; i < i0 + 128; ++i) {
        const float d = T0[(size_t)i * N + o] - T1[(size_t)i * N + o] + PDIST_EPS;
        s += d * d;
    }
    atomicAdd(&colacc[o], s);
}

// Stage 2: sqrt + sum into out[1]
__global__ __launch_bounds__(256) void loss_sqrt_kernel(const float* __restrict__ colacc,
                                                        float* __restrict__ out) {
    const int o = blockIdx.x * 256 + threadIdx.x;    // 0..2047
    float v = sqrtf(colacc[o]);
    v = wave_sum(v);
    if ((threadIdx.x & 31) == 0) atomicAdd(&out[1], v);
}

// ---------------------------------------------------------------------------
// init / finalize
// ---------------------------------------------------------------------------
__global__ __launch_bounds__(256) void init_zero_kernel(float* __restrict__ out,
                                                        float* __restrict__ colacc) {
    const int t = blockIdx.x * 256 + threadIdx.x;
    if (t < 4) out[t] = 0.f;
    if (t < N) colacc[t] = 0.f;
}

__global__ void finalize_kernel(float* __restrict__ out) {
    out[0] = out[1] + out[2] + out[3];
}

// ---------------------------------------------------------------------------
// launch
// ---------------------------------------------------------------------------
extern "C" void kernel_launch(void* const* d_in, const int* in_sizes, int n_in,
                              void* d_out, int out_size, void* d_ws, size_t ws_size,
                              hipStream_t stream) {
    const float* W  = (const float*)d_in[0];   // layer_weights (8,2048,2048) f32
    const float* Pi = (const float*)d_in[1];   // pi_li         (8,2048,2048) f32
    float* out = (float*)d_out;                // [post_loss, loss, col_sq, row_sq]

    const size_t p_bytes = (size_t)NMODELS * N * N * sizeof(float);   // 128 MiB
    const size_t t_bytes = (size_t)2 * N * N * sizeof(float);         //  32 MiB

    float* p      = (float*)d_ws;                                // Sinkhorn work copy
    float* T      = (float*)((char*)d_ws + p_bytes);             // transported[0..1]
    float* colacc = (float*)((char*)d_ws + p_bytes + t_bytes);   // 2048 floats

    // Inputs must stay pristine: iterate on a workspace copy of pi_li.
    hipMemcpyAsync(p, Pi, p_bytes, hipMemcpyDeviceToDevice, stream);

    // 100 Sinkhorn iterations: rows (axis=2) then columns (axis=1).
    for (int it = 0; it < SINKHORN_ITERS; ++it) {
        row_lse_kernel<<<NMODELS * N / 8, 256, 0, stream>>>(p);
        col_lse_kernel<<<NMODELS * N / 256, 256, 0, stream>>>(p);
    }

    init_zero_kernel<<<(N + 255) / 256, 256, 0, stream>>>(out, colacc);

    // marginal penalties over all 8 models
    row_sums_kernel<<<NMODELS * N / 8, 256, 0, stream>>>(p, out);
    col_sums_kernel<<<NMODELS * N / 256, 256, 0, stream>>>(p, out);

    // transported = W @ exp(p) for models 0,1 only (others are dead code)
    dim3 ggrid(N / BN, N / BM, 2);
    gemm_wmma_kernel<<<ggrid, 256, 0, stream>>>(W, p, T);

    // pairwise-distance loss between transported[0].T and transported[1].T
    dim3 lgrid(N / 256, 16);
    loss_partial_kernel<<<lgrid, 256, 0, stream>>>(T, colacc);
    loss_sqrt_kernel<<<N / 256, 256, 0, stream>>>(colacc, out);

    finalize_kernel<<<1, 1, 0, stream>>>(out);
}